// GaussianKernels_2199023255872
// MI455X (gfx1250) — compile-verified
//
#include <hip/hip_runtime.h>
#include <math.h>

// ---------------------------------------------------------------------------
// GaussianKernels for MI455X (gfx1250): fp32 WMMA distance GEMM + radix top-k
// ---------------------------------------------------------------------------

typedef float v2f __attribute__((ext_vector_type(2)));
typedef float v8f __attribute__((ext_vector_type(8)));

#define DDIM 512
#define NCLS 1000
#define KNN 200
#define GAUSS_CONST (1.0f / (2.0f * 10.0f * 10.0f))
#define TILE_COLS 128
#define LDS_STRIDE (DDIM + 4) /* pad 4 floats -> conflict-free ds_load_b64 */

// -------------------------------------------------------------------- norms
__global__ __launch_bounds__(256) void gk_norms_kernel(
    const float* __restrict__ feat, const float* __restrict__ cen,
    float* __restrict__ c2, float* __restrict__ f2, int C, int B) {
  const int wv = threadIdx.x >> 5, lane = threadIdx.x & 31;
  const int row = blockIdx.x * 8 + wv;
  const float* src;
  float* dst;
  if (row < C) {
    src = cen + (size_t)row * DDIM;
    dst = c2 + row;
  } else if (row < C + B) {
    src = feat + (size_t)(row - C) * DDIM;
    dst = f2 + (row - C);
  } else {
    return;
  }
  float s = 0.f;
  for (int d = lane; d < DDIM; d += 32) {
    float v = src[d];
    s = fmaf(v, v, s);
  }
  for (int off = 16; off > 0; off >>= 1) s += __shfl_xor(s, off, 32);
  if (lane == 0) *dst = s;
}

// --------------------------------------------------------- WMMA distance GEMM
// Block: 256 threads (8 waves). Stages a 128-column slab of centres into LDS
// once (centres stream from HBM exactly once), then every wave computes its
// 16-column tile against all 512 feature rows with V_WMMA_F32_16X16X4_F32.
__global__ __launch_bounds__(256) void gk_dist_gemm_kernel(
    const float* __restrict__ F, const float* __restrict__ Cen,
    const float* __restrict__ f2, const float* __restrict__ c2,
    float* __restrict__ d2, int B, int C) {
  extern __shared__ float ldsB[];  // TILE_COLS * LDS_STRIDE floats (~258 KB)
  const int tid = threadIdx.x;
  const int wv = tid >> 5;
  const int lane = tid & 31;
  const int n0blk = blockIdx.x * TILE_COLS;

  // Cooperative stage: centres[n0blk .. n0blk+127][0..511] -> LDS, float4.
  for (int idx = tid; idx < TILE_COLS * (DDIM / 4); idx += 256) {
    const int col = idx >> 7;            // DDIM/4 == 128 float4 per column
    const int k4 = (idx & 127) << 2;
    const int n = n0blk + col;
    float4 v = make_float4(0.f, 0.f, 0.f, 0.f);
    if (n < C) v = *(const float4*)(Cen + (size_t)n * DDIM + k4);
    *(float4*)(&ldsB[col * LDS_STRIDE + k4]) = v;
  }
  __syncthreads();

  const int n0 = n0blk + wv * 16;  // this wave's 16-column tile
  if (n0 + 16 > C) return;         // wave-uniform: EXEC stays all-ones

  const int nl = lane & 15;  // N (and A-row M) index within tile
  const int kh = lane >> 4;  // K-half: lanes 0-15 -> K={0,1}, 16-31 -> K={2,3}
  const float c2n = c2[n0 + nl];
  const float* bbase = &ldsB[(wv * 16 + nl) * LDS_STRIDE + 2 * kh];

  for (int r = 0; r < B / 16; ++r) {
    const int m0 = r * 16;
    const float* abase = F + (size_t)(m0 + nl) * DDIM + 2 * kh;
    v8f acc = {0.f, 0.f, 0.f, 0.f, 0.f, 0.f, 0.f, 0.f};
#pragma unroll 4
    for (int kk = 0; kk < DDIM; kk += 4) {
      v2f a = *(const v2f*)(abase + kk);  // A 16x4 f32 fragment (global, L2-hot)
      v2f b = *(const v2f*)(bbase + kk);  // B 4x16 f32 fragment (LDS)
      acc = __builtin_amdgcn_wmma_f32_16x16x4_f32(
          false, a, false, b, (short)0, acc, false, false);
    }
    // C/D layout: VGPR i holds M = m0 + kh*8 + i, N = n0 + nl.
    const int mrow = m0 + kh * 8;
    const size_t ob = (size_t)mrow * (size_t)C + (size_t)(n0 + nl);
#pragma unroll
    for (int i = 0; i < 8; ++i) {
      d2[ob + (size_t)i * C] = f2[mrow + i] + c2n - 2.0f * acc[i];
    }
  }
}

// --------------------------------------------------- top-k select + classify
__device__ __forceinline__ unsigned gk_mapf(float v) {
  unsigned x = __float_as_uint(v);
  unsigned m = (x & 0x80000000u) ? 0xFFFFFFFFu : 0x80000000u;
  return x ^ m;  // monotone float -> uint
}

__global__ __launch_bounds__(256) void gk_select_kernel(
    const float* __restrict__ d2, const float* __restrict__ weight,
    const int* __restrict__ labels, float* __restrict__ out, int C) {
  __shared__ unsigned hist[256];
  __shared__ float p[NCLS];
  __shared__ float wsum[8];
  __shared__ unsigned s_pref, s_need, s_ticket;
  __shared__ float s_logsum;

  const int tid = threadIdx.x;
  const float* row = d2 + (size_t)blockIdx.x * C;
  if (tid == 0) {
    s_pref = 0u;
    s_need = KNN;
    s_ticket = 0u;
  }

  // 4-pass MSD radix select: exact 200th-smallest mapped value.
  for (int pass = 0; pass < 4; ++pass) {
    const int shift = 24 - 8 * pass;
    const unsigned pmask = (pass == 0) ? 0u : (0xFFFFFFFFu << (shift + 8));
    hist[tid] = 0u;
    __syncthreads();
    const unsigned pref = s_pref;
    for (int i = tid; i < C; i += 256) {
      const unsigned u = gk_mapf(row[i]);
      if ((u & pmask) == pref) atomicAdd(&hist[(u >> shift) & 255u], 1u);
    }
    __syncthreads();
    if (tid == 0) {
      unsigned need = s_need, cum = 0u;
      int sel = 255;
      for (int j = 0; j < 256; ++j) {
        const unsigned h = hist[j];
        if (cum + h >= need) { sel = j; break; }
        cum += h;
      }
      s_pref = pref | ((unsigned)sel << shift);
      s_need = need - cum;  // remaining rank inside the selected bucket
    }
    __syncthreads();
  }
  const unsigned T = s_pref;       // mapped value of the K-th smallest
  const unsigned takeEq = s_need;  // how many ==T entries belong to top-K

  for (int j = tid; j < NCLS; j += 256) p[j] = 0.f;
  __syncthreads();

  // Accumulate Gaussian kernel weights of the top-K neighbours per class.
  for (int i = tid; i < C; i += 256) {
    const float v = row[i];
    const unsigned u = gk_mapf(v);
    bool take = (u < T);
    if (!take && u == T) take = (atomicAdd(&s_ticket, 1u) < takeEq);
    if (take) {
      const float d = expf(-v * GAUSS_CONST) * expf(weight[i]);
      atomicAdd(&p[labels[i]], d);
    }
  }
  __syncthreads();

  // log-softmax style finish: p = where(p==0, 1e-10, p); out = log(p/sum(p)).
  float partial = 0.f;
  for (int j = tid; j < NCLS; j += 256) {
    float pj = p[j];
    if (pj == 0.f) pj = 1e-10f;
    partial += pj;
  }
  for (int off = 16; off > 0; off >>= 1) partial += __shfl_xor(partial, off, 32);
  if ((tid & 31) == 0) wsum[tid >> 5] = partial;
  __syncthreads();
  if (tid == 0) {
    float t = 0.f;
    for (int w = 0; w < 8; ++w) t += wsum[w];
    s_logsum = logf(t);
  }
  __syncthreads();
  const float ls = s_logsum;
  float* orow = out + (size_t)blockIdx.x * NCLS;
  for (int j = tid; j < NCLS; j += 256) {
    float pj = p[j];
    if (pj == 0.f) pj = 1e-10f;
    orow[j] = logf(pj) - ls;
  }
}

// --------------------------------------------------------------------- launch
extern "C" void kernel_launch(void* const* d_in, const int* in_sizes, int n_in,
                              void* d_out, int out_size, void* d_ws,
                              size_t ws_size, hipStream_t stream) {
  const float* features = (const float*)d_in[0];  // [B, 512]
  const float* centres = (const float*)d_in[1];   // [C, 512]
  const int* labels = (const int*)d_in[2];        // [C]
  const float* weight = (const float*)d_in[3];    // [C]
  float* out = (float*)d_out;                     // [B, 1000]

  const int B = in_sizes[0] / DDIM;  // 512
  const int C = in_sizes[2];         // 100000

  // Workspace layout: d2 matrix, then c2, then f2 (256B aligned).
  char* ws = (char*)d_ws;
  float* d2 = (float*)ws;
  const size_t d2bytes = ((size_t)B * (size_t)C * sizeof(float) + 255) & ~(size_t)255;
  float* c2 = (float*)(ws + d2bytes);
  float* f2 = c2 + (((size_t)C + 63) & ~(size_t)63);

  // 1) row norms (one wave per row, 8 rows per block)
  {
    const int rows = C + B;
    gk_norms_kernel<<<(rows + 7) / 8, 256, 0, stream>>>(features, centres, c2,
                                                        f2, C, B);
  }
  // 2) fp32 WMMA distance GEMM, centres staged once into 258KB of LDS
  {
    const int colTiles = C / 16;               // 6250
    const int blocks = (colTiles + 7) / 8;     // 8 col-tiles per block
    const size_t ldsBytes = (size_t)TILE_COLS * LDS_STRIDE * sizeof(float);
    gk_dist_gemm_kernel<<<blocks, 256, ldsBytes, stream>>>(features, centres,
                                                           f2, c2, d2, B, C);
  }
  // 3) exact per-row top-200 radix select + class accumulation + log-normalize
  gk_select_kernel<<<B, 256, 0, stream>>>(d2, weight, labels, out, C);
}